// ToRGB_43147241455832
// MI455X (gfx1250) — compile-verified
//
#include <hip/hip_runtime.h>

typedef __attribute__((ext_vector_type(2))) float v2f;
typedef __attribute__((ext_vector_type(8))) float v8f;

#define IN_CH   128
#define STYLE   512
#define OUT_CH  3
#define HW      65536     // 256*256
#define SK      128       // skip spatial size

// Kernel 1: s[b,c] = mod_bias[c] + (1/sqrt(512)) * sum_k style[b,k]*mod_w[c,k]
__global__ __launch_bounds__(128) void torgb_style_kernel(
    const float* __restrict__ style, const float* __restrict__ modw,
    const float* __restrict__ modb, float* __restrict__ s_out)
{
    const int b = blockIdx.x;
    const int c = threadIdx.x;
    const float mod_scale = 0.04419417382415922f;  // 1/sqrt(512)
    const float* st = style + b * STYLE;
    const float* mw = modw + (size_t)c * STYLE;
    float sum = 0.f;
#pragma unroll 8
    for (int k = 0; k < STYLE; ++k) sum += st[k] * mw[k];
    s_out[b * IN_CH + c] = sum * mod_scale + modb[c];
}

// Kernel 2: streaming modulated 1x1 conv via V_WMMA_F32_16X16X4_F32 + blur-upsampled skip.
// A (16x4)  = Weff (M=16 padded output channels, K=4 input channels)
// B (4x16)  = x pixels (K=4 channels, N=16 consecutive pixels)
// D (16x16) = 16 out-channels x 16 pixels; rows 0..2 are the RGB result.
__global__ __launch_bounds__(256) void torgb_main_kernel(
    const float* __restrict__ x, const float* __restrict__ skip,
    const float* __restrict__ cw, const float* __restrict__ bias,
    const float* __restrict__ s, float* __restrict__ out)
{
    __shared__ float weff[IN_CH * 16];   // [c][o] , o padded 3 -> 16 with zeros (8 KB)
    const int tid = threadIdx.x;
    const int b = blockIdx.y;
    const float conv_scale = 0.08838834764831845f;  // 1/sqrt(128)

    for (int idx = tid; idx < IN_CH * 16; idx += 256) {
        const int c = idx >> 4, o = idx & 15;
        weff[idx] = (o < OUT_CH) ? conv_scale * cw[o * IN_CH + c] * s[b * IN_CH + c]
                                 : 0.f;
    }
    __syncthreads();

    const int lane  = tid & 31;
    const int wave  = tid >> 5;
    const int n     = lane & 15;       // N index (pixel within tile / M row of A)
    const int khalf = lane >> 4;       // selects K pair {0,1} vs {2,3}
    const int pwave = blockIdx.x * 512 + wave * 64;   // 4 tiles of 16 pixels per wave
    const float* xb = x + (size_t)b * IN_CH * HW;

    v8f acc[4] = {};                   // 4 accumulator tiles (64 pixels / wave)

#pragma unroll 4
    for (int ci = 0; ci < IN_CH; ci += 4) {
        const int k0 = ci + khalf * 2;
        // A: 16x4 f32 layout — lanes 0-15: v0=K0,v1=K1 ; lanes 16-31: v0=K2,v1=K3
        v2f a;
        a.x = weff[k0 * 16 + n];
        a.y = weff[(k0 + 1) * 16 + n];
        const float* xp = xb + (size_t)k0 * HW + pwave + n;
        // B: 4x16 f32, rows striped across lanes per VGPR (v0: K0/K2, v1: K1/K3)
#pragma unroll
        for (int t = 0; t < 4; ++t) {
            v2f bm;
            bm.x = xp[t * 16];
            bm.y = xp[t * 16 + HW];
            acc[t] = __builtin_amdgcn_wmma_f32_16x16x4_f32(
                false, a, false, bm, (short)0, acc[t], false, false);
        }
    }

    // Epilogue: D rows M=0..2 live in acc[t][0..2] of lanes 0-15 (N=pixel).
    if (lane < 16) {
        const float bz0 = bias[0], bz1 = bias[1], bz2 = bias[2];
        const float bz[3] = { bz0, bz1, bz2 };
#pragma unroll
        for (int t = 0; t < 4; ++t) {
            const int p = pwave + t * 16 + n;
            const int y = p >> 8, xc = p & 255;
            // upfirdn2d(up=2, pad=(2,1), k=[1,3,3,1]/4 per axis) -> 2 taps/axis
            int iy0, iy1, ix0, ix1;
            float wy0, wy1, wx0, wx1;
            if (y & 1) { iy0 = y >> 1;  wy0 = 0.75f; iy1 = iy0 + 1; wy1 = 0.25f; }
            else       { iy1 = y >> 1;  wy1 = 0.75f; iy0 = iy1 - 1; wy0 = 0.25f; }
            if (iy0 < 0)      { iy0 = 0;      wy0 = 0.f; }
            if (iy1 > SK - 1) { iy1 = SK - 1; wy1 = 0.f; }
            if (xc & 1) { ix0 = xc >> 1; wx0 = 0.75f; ix1 = ix0 + 1; wx1 = 0.25f; }
            else        { ix1 = xc >> 1; wx1 = 0.75f; ix0 = ix1 - 1; wx0 = 0.25f; }
            if (ix0 < 0)      { ix0 = 0;      wx0 = 0.f; }
            if (ix1 > SK - 1) { ix1 = SK - 1; wx1 = 0.f; }
#pragma unroll
            for (int o = 0; o < OUT_CH; ++o) {
                const float* sk = skip + ((size_t)(b * OUT_CH + o)) * SK * SK;
                const float up =
                    wy0 * (wx0 * sk[iy0 * SK + ix0] + wx1 * sk[iy0 * SK + ix1]) +
                    wy1 * (wx0 * sk[iy1 * SK + ix0] + wx1 * sk[iy1 * SK + ix1]);
                out[((size_t)b * OUT_CH + o) * HW + p] = acc[t][o] + bz[o] + up;
            }
        }
    }
}

extern "C" void kernel_launch(void* const* d_in, const int* in_sizes, int n_in,
                              void* d_out, int out_size, void* d_ws, size_t ws_size,
                              hipStream_t stream) {
    const float* x     = (const float*)d_in[0];   // [8,128,256,256]
    const float* style = (const float*)d_in[1];   // [8,512]
    const float* skip  = (const float*)d_in[2];   // [8,3,128,128]
    const float* cw    = (const float*)d_in[3];   // [3,128]
    const float* mw    = (const float*)d_in[4];   // [128,512]
    const float* mb    = (const float*)d_in[5];   // [128]
    const float* bias  = (const float*)d_in[6];   // [3]
    float* out = (float*)d_out;                   // [8,3,256,256]
    float* s   = (float*)d_ws;                    // [8,128] scratch

    torgb_style_kernel<<<dim3(8), dim3(128), 0, stream>>>(style, mw, mb, s);
    torgb_main_kernel<<<dim3(128, 8), dim3(256), 0, stream>>>(x, skip, cw, bias, s, out);
}